// EntailmentSelfAttention_80221399155145
// MI455X (gfx1250) — compile-verified
//
#include <hip/hip_runtime.h>

typedef __attribute__((ext_vector_type(2))) float v2f;
typedef __attribute__((ext_vector_type(8))) float v8f;

#define NS   16      // N*S
#define LSEQ 512
#define EMB  1024
#define HH   16
#define DD   64
#define SCALE 0.044194173824159216f      // 1/sqrt(512)
#define EMASKV (-1e20f * SCALE)          // masked energy after /sqrt(keylen)
#define NEGBIG (-3.402823466e38f)

// ---------------- CDNA5 async global->LDS (ASYNCcnt path) -------------------
// VDST = VGPR with wave-relative LDS byte address (generic LDS addr low bits),
// VADDR = 64-bit global address, saddr disabled ("off").
__device__ __forceinline__ void async_ld_b128(void* lds, const void* gaddr) {
  unsigned int l = (unsigned int)(unsigned long long)lds;  // LDS aperture: addr[31:0]
  asm volatile("global_load_async_to_lds_b128 %0, %1, off"
               :: "v"(l), "v"(gaddr) : "memory");
}
#define S_WAIT_ASYNCCNT(n) asm volatile("s_wait_asynccnt " #n ::: "memory")

// ---------------- WMMA helpers (V_WMMA_F32_16X16X4_F32, full fp32) ----------
__device__ __forceinline__ v8f wmma4(v2f a, v2f b, v8f c) {
  return __builtin_amdgcn_wmma_f32_16x16x4_f32(
      false, a, false, b, (short)0, c, false, false);
}
// A fragment from row-major A[16][lda] in LDS (M = lane&15, K pair by half)
__device__ __forceinline__ v2f frag_a(const float* A, int lda, int k0, int lane) {
  int m  = lane & 15;
  int kk = k0 + ((lane >> 4) << 1);
  v2f a; a.x = A[m * lda + kk]; a.y = A[m * lda + kk + 1]; return a;
}
// B fragment from row-major B[K][ldb] (B[k][n]); N = lane&15
__device__ __forceinline__ v2f frag_b(const float* B, int ldb, int k0, int lane) {
  int n  = lane & 15;
  int kk = k0 + ((lane >> 4) << 1);
  v2f b; b.x = B[kk * ldb + n]; b.y = B[(kk + 1) * ldb + n]; return b;
}
// B fragment for B = W^T, i.e. B[k][n] = W[n][k] with W row-major [n][ldw]
__device__ __forceinline__ v2f frag_bT(const float* W, int ldw, int k0, int lane) {
  int n  = lane & 15;
  int kk = k0 + ((lane >> 4) << 1);
  v2f b; b.x = W[n * ldw + kk]; b.y = W[n * ldw + kk + 1]; return b;
}

// ---------------- Kernel 1: Q/K/V head projections (X @ W^T) ----------------
__global__ __launch_bounds__(128)
void proj_kernel(const float* __restrict__ values, const float* __restrict__ keys,
                 const float* __restrict__ query,
                 const float* __restrict__ Wv, const float* __restrict__ Wk,
                 const float* __restrict__ Wq,
                 float* __restrict__ Vp, float* __restrict__ Kp, float* __restrict__ Qp) {
  __shared__ float Xs[64 * 64];
  __shared__ float Ws[64 * 64];
  const int tid  = threadIdx.x;
  const int lane = tid & 31;
  const int wave = tid >> 5;
  const int l0 = blockIdx.x * 64;
  const int h  = blockIdx.y;
  const int ns = blockIdx.z;
  const int n    = lane & 15;
  const int half = lane >> 4;

  const float* xin[3]  = {values, keys, query};
  const float* win[3]  = {Wv, Wk, Wq};
  float*       outp[3] = {Vp, Kp, Qp};

  for (int p = 0; p < 3; ++p) {
    const float* src = xin[p];
    for (int t = 0; t < 8; ++t) {                    // 64x64 X chunk, async
      int idx = tid + t * 128;
      int row = idx >> 4, cv = idx & 15;
      async_ld_b128(&Xs[row * 64 + cv * 4],
                    &src[((size_t)(ns * LSEQ + l0 + row)) * EMB + h * DD + cv * 4]);
    }
    for (int t = 0; t < 8; ++t) {                    // 64x64 weight, async
      int idx = tid + t * 128;
      async_ld_b128(&Ws[idx * 4], &win[p][idx * 4]);
    }
    S_WAIT_ASYNCCNT(0);
    __syncthreads();

    const float* Arow = &Xs[(wave * 16) * 64];
    v8f zero = {};
    v8f acc[4] = {zero, zero, zero, zero};
    for (int ct = 0; ct < 4; ++ct) {
      const float* Wn = &Ws[(ct * 16) * 64];
      for (int k0 = 0; k0 < 64; k0 += 4)
        acc[ct] = wmma4(frag_a(Arow, 64, k0, lane), frag_bT(Wn, 64, k0, lane), acc[ct]);
    }
    // stage C tiles in own wave's Xs region -> coalesced float4 stores
    float* stage = &Xs[(wave * 16) * 64];
    for (int ct = 0; ct < 4; ++ct)
      for (int i = 0; i < 8; ++i) {
        int m = i + half * 8;
        stage[m * 64 + ct * 16 + n] = acc[ct][i];
      }
    float* dst = outp[p];
    size_t base = (size_t)(ns * HH + h) * LSEQ + l0 + wave * 16;
    for (int t = 0; t < 8; ++t) {
      int idx = lane + t * 32;                       // 256 vec4 per wave region
      int row = idx >> 4, cv = idx & 15;
      *(float4*)(&dst[(base + row) * DD + cv * 4]) =
          *(const float4*)(&stage[row * 64 + cv * 4]);
    }
    __syncthreads();
  }
}

// ---------------- Kernel 2: column-wise (over q) online softmax stats -------
__global__ __launch_bounds__(128)
void stats_kernel(const float* __restrict__ Qp, const float* __restrict__ Kp,
                  const int* __restrict__ mask,
                  float* __restrict__ Mbuf, float* __restrict__ Zbuf) {
  __shared__ float Ks[64 * 64];
  __shared__ float Qs[16 * 64];
  __shared__ int   ms[16];
  const int tid  = threadIdx.x;
  const int lane = tid & 31;
  const int wave = tid >> 5;
  const int c0 = blockIdx.x * 64;
  const int h  = blockIdx.y;
  const int ns = blockIdx.z;
  const int nsh  = ns * HH + h;
  const int n    = lane & 15;
  const int half = lane >> 4;

  const float* Kbase = &Kp[((size_t)nsh * LSEQ + c0) * DD];
  for (int t = 0; t < 8; ++t) {
    int idx = tid + t * 128;
    async_ld_b128(&Ks[idx * 4], &Kbase[idx * 4]);
  }

  float rM = NEGBIG, rS = 0.f;
  const float* Bn = &Ks[(wave * 16) * 64];

  for (int qt = 0; qt < 32; ++qt) {
    int q0 = qt * 16;
    __syncthreads();
    for (int t = 0; t < 2; ++t) {
      int idx = tid + t * 128;
      async_ld_b128(&Qs[idx * 4], &Qp[((size_t)nsh * LSEQ + q0) * DD + idx * 4]);
    }
    if (tid < 16) ms[tid] = mask[ns * LSEQ + q0 + tid];
    S_WAIT_ASYNCCNT(0);
    __syncthreads();

    v8f c = {};
    for (int k0 = 0; k0 < 64; k0 += 4)
      c = wmma4(frag_a(Qs, 64, k0, lane), frag_bT(Bn, 64, k0, lane), c);

    float vals[8];
    for (int i = 0; i < 8; ++i) {
      int m = i + half * 8;
      vals[i] = ms[m] ? c[i] * SCALE : EMASKV;
    }
    float tM = vals[0];
    for (int i = 1; i < 8; ++i) tM = fmaxf(tM, vals[i]);
    tM = fmaxf(tM, __shfl_xor(tM, 16, 32));
    float nM = fmaxf(rM, tM);
    float tS = 0.f;
    for (int i = 0; i < 8; ++i) tS += __expf(vals[i] - nM);
    tS += __shfl_xor(tS, 16, 32);
    rS = rS * __expf(rM - nM) + tS;
    rM = nM;
  }
  if (half == 0) {
    int col = c0 + wave * 16 + n;
    Mbuf[(size_t)nsh * LSEQ + col] = rM;
    Zbuf[(size_t)nsh * LSEQ + col] = rS;
  }
}

// ---------------- Kernel 3: recompute E, P = exp(E-M)/Z, O += P @ V ---------
__global__ __launch_bounds__(128)
void attnout_kernel(const float* __restrict__ Qp, const float* __restrict__ Kp,
                    const float* __restrict__ Vp, const int* __restrict__ mask,
                    const float* __restrict__ Mbuf, const float* __restrict__ Zbuf,
                    float* __restrict__ Hout) {
  __shared__ float Qs[64 * 64];
  __shared__ float Ks[16 * 64];
  __shared__ float Vs[16 * 64];
  __shared__ float Ps[4][16 * 16];
  __shared__ float Ml[16];
  __shared__ float Zl[16];
  __shared__ int   ms[64];
  const int tid  = threadIdx.x;
  const int lane = tid & 31;
  const int wave = tid >> 5;
  const int q0 = blockIdx.x * 64;
  const int h  = blockIdx.y;
  const int ns = blockIdx.z;
  const int nsh  = ns * HH + h;
  const int n    = lane & 15;
  const int half = lane >> 4;

  for (int t = 0; t < 8; ++t) {
    int idx = tid + t * 128;
    async_ld_b128(&Qs[idx * 4], &Qp[((size_t)nsh * LSEQ + q0) * DD + idx * 4]);
  }
  if (tid < 64) ms[tid] = mask[ns * LSEQ + q0 + tid];

  v8f zero = {};
  v8f acc[4] = {zero, zero, zero, zero};
  const float* Arow = &Qs[(wave * 16) * 64];

  for (int lt = 0; lt < 32; ++lt) {
    int l0 = lt * 16;
    __syncthreads();
    for (int t = 0; t < 2; ++t) {
      int idx = tid + t * 128;
      async_ld_b128(&Ks[idx * 4], &Kp[((size_t)nsh * LSEQ + l0) * DD + idx * 4]);
      async_ld_b128(&Vs[idx * 4], &Vp[((size_t)nsh * LSEQ + l0) * DD + idx * 4]);
    }
    if (tid < 16) {
      Ml[tid] = Mbuf[(size_t)nsh * LSEQ + l0 + tid];
      Zl[tid] = Zbuf[(size_t)nsh * LSEQ + l0 + tid];
    }
    S_WAIT_ASYNCCNT(0);
    __syncthreads();

    v8f c = {};
    for (int k0 = 0; k0 < 64; k0 += 4)
      c = wmma4(frag_a(Arow, 64, k0, lane), frag_bT(Ks, 64, k0, lane), c);

    float invZ = 1.f / Zl[n];
    float Mc   = Ml[n];
    for (int i = 0; i < 8; ++i) {
      int m = i + half * 8;
      float val = ms[wave * 16 + m] ? c[i] * SCALE : EMASKV;
      Ps[wave][m * 16 + n] = __expf(val - Mc) * invZ;
    }
    for (int ct = 0; ct < 4; ++ct)
      for (int k0 = 0; k0 < 16; k0 += 4)
        acc[ct] = wmma4(frag_a(&Ps[wave][0], 16, k0, lane),
                        frag_b(&Vs[ct * 16], 64, k0, lane), acc[ct]);
  }

  // stage O (16x64) in own wave's Qs region -> coalesced float4 stores
  float* stage = (float*)&Qs[(wave * 16) * 64];
  for (int ct = 0; ct < 4; ++ct)
    for (int i = 0; i < 8; ++i) {
      int m = i + half * 8;
      stage[m * 64 + ct * 16 + n] = acc[ct][i];
    }
  for (int t = 0; t < 8; ++t) {
    int idx = lane + t * 32;
    int row = idx >> 4, cv = idx & 15;
    size_t grow = (size_t)ns * LSEQ + q0 + wave * 16 + row;
    *(float4*)(&Hout[grow * EMB + h * DD + cv * 4]) =
        *(const float4*)(&stage[row * 64 + cv * 4]);
  }
}

// ---------------- Kernel 4: Y = Hout @ Wo^T + bo  (8192x1024x1024) ----------
// Double-buffered async-LDS pipeline: each prefetch = 16 wave-level async ops.
__global__ __launch_bounds__(128)
void outproj_kernel(const float* __restrict__ Hout, const float* __restrict__ Wo,
                    const float* __restrict__ bo, float* __restrict__ Y) {
  __shared__ float As[2][64 * 64];
  __shared__ float Bs[2][64 * 64];
  const int tid  = threadIdx.x;
  const int lane = tid & 31;
  const int wave = tid >> 5;
  const int r0 = blockIdx.x * 64;
  const int c0 = blockIdx.y * 64;
  const int n    = lane & 15;
  const int half = lane >> 4;

  auto prefetch = [&](int kt, int buf) {
    int k0g = kt * 64;
    for (int t = 0; t < 8; ++t) {
      int idx = tid + t * 128;
      int row = idx >> 4, cv = idx & 15;
      async_ld_b128(&As[buf][row * 64 + cv * 4],
                    &Hout[((size_t)(r0 + row)) * EMB + k0g + cv * 4]);
      async_ld_b128(&Bs[buf][row * 64 + cv * 4],
                    &Wo[((size_t)(c0 + row)) * EMB + k0g + cv * 4]);
    }
  };

  v8f zero = {};
  v8f acc[4] = {zero, zero, zero, zero};

  prefetch(0, 0);
  for (int kt = 0; kt < 16; ++kt) {
    int buf = kt & 1;
    if (kt < 15) {
      prefetch(kt + 1, buf ^ 1);
      S_WAIT_ASYNCCNT(16);     // <=16 outstanding => current buffer complete
    } else {
      S_WAIT_ASYNCCNT(0);
    }
    __syncthreads();
    const float* Arow = &As[buf][(wave * 16) * 64];
    for (int ct = 0; ct < 4; ++ct) {
      const float* Wn = &Bs[buf][(ct * 16) * 64];
      for (int k0 = 0; k0 < 64; k0 += 4)
        acc[ct] = wmma4(frag_a(Arow, 64, k0, lane), frag_bT(Wn, 64, k0, lane), acc[ct]);
    }
    __syncthreads();           // all waves done with buf before it is refilled
  }

  // stage (with bias) in own wave's As region -> coalesced float4 stores
  float* stage = &As[0][(wave * 16) * 64];
  for (int ct = 0; ct < 4; ++ct) {
    float bias = bo[c0 + ct * 16 + n];
    for (int i = 0; i < 8; ++i) {
      int m = i + half * 8;
      stage[m * 64 + ct * 16 + n] = acc[ct][i] + bias;
    }
  }
  for (int t = 0; t < 8; ++t) {
    int idx = lane + t * 32;
    int row = idx >> 4, cv = idx & 15;
    *(float4*)(&Y[((size_t)(r0 + wave * 16 + row)) * EMB + c0 + cv * 4]) =
        *(const float4*)(&stage[row * 64 + cv * 4]);
  }
}

// ---------------------------------------------------------------------------
extern "C" void kernel_launch(void* const* d_in, const int* in_sizes, int n_in,
                              void* d_out, int out_size, void* d_ws, size_t ws_size,
                              hipStream_t stream) {
  const float* values = (const float*)d_in[0];
  const float* keys   = (const float*)d_in[1];
  const float* query  = (const float*)d_in[2];
  const int*   mask   = (const int*)d_in[3];
  const float* Wv = (const float*)d_in[4];
  const float* Wk = (const float*)d_in[5];
  const float* Wq = (const float*)d_in[6];
  const float* Wo = (const float*)d_in[7];
  const float* bo = (const float*)d_in[8];
  float* Y = (float*)d_out;

  const size_t per   = (size_t)NS * HH * LSEQ * DD;
  const size_t stats = (size_t)NS * HH * LSEQ;
  float* ws   = (float*)d_ws;
  float* Vp   = ws;
  float* Kp   = Vp + per;
  float* Qp   = Kp + per;
  float* Mbuf = Qp + per;
  float* Zbuf = Mbuf + stats;
  float* Hout = Zbuf + stats;

  dim3 blk(128);
  proj_kernel   <<<dim3(LSEQ / 64, HH, NS), blk, 0, stream>>>(values, keys, query,
                                                              Wv, Wk, Wq, Vp, Kp, Qp);
  stats_kernel  <<<dim3(LSEQ / 64, HH, NS), blk, 0, stream>>>(Qp, Kp, mask, Mbuf, Zbuf);
  attnout_kernel<<<dim3(LSEQ / 64, HH, NS), blk, 0, stream>>>(Qp, Kp, Vp, mask,
                                                              Mbuf, Zbuf, Hout);
  outproj_kernel<<<dim3((NS * LSEQ) / 64, EMB / 64, 1), blk, 0, stream>>>(Hout, Wo, bo, Y);
}